// NuFFT_34162169872608
// MI455X (gfx1250) — compile-verified
//
#include <hip/hip_runtime.h>
#include <hip/hip_bf16.h>
#include <math.h>

// ---------------- problem constants ----------------
#define NPIX  1024
#define NCHAN 4
#define NVIS  200000
#define GSZ   2048        // OSF * NPIX
#define JW    6
#define PI_F  3.14159265358979323846f

// 2048 = N2(32, stage 1, K truncated to 16 by zero padding) * N1(64, stage 2)

typedef float v2f __attribute__((ext_vector_type(2)));
typedef float v8f __attribute__((ext_vector_type(8)));
typedef unsigned int v4u __attribute__((ext_vector_type(4)));
typedef int v4i __attribute__((ext_vector_type(4)));
typedef int v8i __attribute__((ext_vector_type(8)));

#define WMMA_F32(A, B, C) \
  __builtin_amdgcn_wmma_f32_16x16x4_f32(false, (A), false, (B), (short)0, (C), false, false)

__device__ __forceinline__ float betaf() {
  // pi * sqrt((J/OSF)^2 * (OSF-0.5)^2 - 0.8) = pi * sqrt(19.45)
  return PI_F * sqrtf(19.45f);
}

// Abramowitz & Stegun I0 approximation (matches jnp.i0 to ~1e-7 rel)
__device__ __forceinline__ float i0f(float x) {
  float ax = fabsf(x);
  if (ax < 3.75f) {
    float t = ax / 3.75f;
    t *= t;
    return 1.0f + t*(3.5156229f + t*(3.0899424f + t*(1.2067492f +
               t*(0.2659732f + t*(0.0360768f + t*0.0045813f)))));
  }
  float t = 3.75f / ax;
  float p = 0.39894228f + t*(0.01328592f + t*(0.00225319f + t*(-0.00157565f +
            t*(0.00916281f + t*(-0.02057706f + t*(0.02635537f +
            t*(-0.01647633f + t*0.00392377f)))))));
  return expf(ax) * rsqrtf(ax) * p;
}

// -------- Tensor Data Mover: DMA one contiguous 1-D line into LDS --------
// Builds a D# descriptor (CDNA5 ISA) for a 1-D tile of
// `elems` elements of size (1<<dszlog) bytes and issues TENSOR_LOAD_TO_LDS.
// Tracked by TENSORcnt; caller must s_wait_tensorcnt before reading LDS.
__device__ __forceinline__ void tdm_load_line(void* lds_dst, const void* gsrc,
                                              unsigned elems, unsigned dszlog) {
  unsigned lds_off = (unsigned)(unsigned long long)(uintptr_t)lds_dst; // LDS byte offset
  unsigned long long ga = (unsigned long long)(uintptr_t)gsrc;

  v4u g0;
  g0.x = 1u;                                        // count=1 valid descriptor
  g0.y = lds_off;                                   // lds_addr (bytes)
  g0.z = (unsigned)(ga & 0xFFFFFFFFu);              // global_addr[31:0]
  g0.w = (unsigned)((ga >> 32) & 0x1FFFFFFu)        // global_addr[56:32]
         | (2u << 30);                              // type = 2 ("image")

  // Group1 bit layout (256b): [17:16]=data_size, [79:48]=tensor_dim0,
  // [111:80]=tensor_dim1, [127:112]=tile_dim0, [143:128]=tile_dim1,
  // [159:144]=tile_dim2, [207:160]=tensor_dim0_stride, [255:208]=dim1_stride
  v8i g1;
  g1[0] = (int)((dszlog == 3u ? 3u : 2u) << 16);    // workgroup_mask=0, data_size
  g1[1] = (int)((elems & 0xFFFFu) << 16);           // tensor_dim0[15:0]
  g1[2] = (int)((elems >> 16) | (1u << 16));        // tensor_dim0[31:16], tensor_dim1=1
  g1[3] = (int)((elems & 0xFFFFu) << 16);           // tensor_dim1 hi=0, tile_dim0
  g1[4] = 0;                                        // tile_dim1=0 (1-D), tile_dim2=0
  g1[5] = (int)elems;                               // tensor_dim0_stride[31:0]
  g1[6] = 0;                                        // stride0 hi, dim1_stride lo
  g1[7] = 0;

  v4i g2 = {1, 0, 0, 0};                            // tensor_dim2=1, rest unused
  v4i g3 = {0, 0, 0, 0};
  v8i g4 = {0, 0, 0, 0, 0, 0, 0, 0};                // extra group (clang-23 form)

  // amdgpu-toolchain (clang-23) 6-arg form:
  // (uint32x4, int32x8, int32x4, int32x4, int32x8, i32 cpol)
  __builtin_amdgcn_tensor_load_to_lds(g0, g1, g2, g3, g4, 0);
}

// ---------------- workspace layout (bytes) ----------------
// [0)      inv_s1  : 1024 f32                       (4 KB)
// [4096)   tbl_d32 : 32x16 float2  DFT32 (K<16)     (4 KB)
// [8192)   tbl_d64 : 64x64 float2  DFT64            (32 KB)
// [40960)  tbl_tw  : 64x32 float2  W2048^(n1*k2)    (16 KB)
// [57344)  R1t     : [c][ku][y]  4*2048*1024 float2 (64 MB)
// [+64MB)  GHAT_T  : [c][ku][kv] 4*2048*2048 float2 (128 MB)
#define WSO_INVS1 ((size_t)0)
#define WSO_D32   ((size_t)4096)
#define WSO_D64   ((size_t)8192)
#define WSO_TW    ((size_t)40960)
#define WSO_R1T   ((size_t)57344)
#define WSO_GHAT  (WSO_R1T + (size_t)NCHAN * GSZ * NPIX * sizeof(float2))

// ---------------- table init ----------------
__global__ void nufft_init_tables(float* __restrict__ invs1,
                                  float2* __restrict__ d32,
                                  float2* __restrict__ d64,
                                  float2* __restrict__ twd) {
  int t = blockIdx.x * blockDim.x + threadIdx.x;
  int stride = blockDim.x * gridDim.x;
  float beta = betaf();
  for (int i = t; i < NPIX; i += stride) {
    float f  = (float)i / (float)GSZ;
    float z  = PI_F * (float)JW * f;
    float s  = sqrtf(beta * beta - z * z);       // z < beta for i < 1024
    float s1 = (float)JW * sinhf(s) / s;
    invs1[i] = 1.0f / s1;
  }
  for (int i = t; i < 32 * 16; i += stride) {    // d32[m*16+k] = W32^(m*k)
    int m = i >> 4, k = i & 15;
    int mk = (m * k) & 31;
    float sn, cs;
    sincosf(-2.0f * PI_F * (float)mk / 32.0f, &sn, &cs);
    d32[i] = make_float2(cs, sn);
  }
  for (int i = t; i < 64 * 64; i += stride) {    // d64[m*64+k] = W64^(m*k)
    int m = i >> 6, k = i & 63;
    int mk = (m * k) & 63;
    float sn, cs;
    sincosf(-2.0f * PI_F * (float)mk / 64.0f, &sn, &cs);
    d64[i] = make_float2(cs, sn);
  }
  for (int i = t; i < 64 * 32; i += stride) {    // tw[n1*32+k2] = W2048^(n1*k2)
    int n1 = i >> 5, k2 = i & 31;
    int mk = (n1 * k2) & (GSZ - 1);
    float sn, cs;
    sincosf(-2.0f * PI_F * (float)mk / (float)GSZ, &sn, &cs);
    twd[i] = make_float2(cs, sn);
  }
}

// ---------------- four-step 2048-pt FFT, one wave per line ----------------
// PASS 1: line=(c,y), real input row of cube (TDM-staged into LDS),
//         apodization folded into B-operand build, output R1t[c][ku][y].
// PASS 2: line=(c,ku), complex input column R1t[c][ku][0..1023] (TDM-staged),
//         output GHAT_T[c][ku][kv] (contiguous in kv).
template <int PASS>
__global__ void __launch_bounds__(32, 1)
nufft_fft2048(const float*  __restrict__ cube,
              const float*  __restrict__ invs1,
              const float2* __restrict__ tD32,
              const float2* __restrict__ tD64,
              const float2* __restrict__ tTW,
              const float2* __restrict__ cin,
              float2*       __restrict__ cout) {
  __shared__ float2 bufIn[NPIX];   // staged input line (pass1 uses first 4 KB)
  __shared__ float2 bufB[GSZ];     // Z[n1][k2] (twiddled stage-1 output)

  const int lane = threadIdx.x;
  const int lo   = lane & 15;      // N / M fast index
  const int lh   = lane >> 4;      // lane half selects K pairs / M high rows

  const int line = blockIdx.x;
  float  invs1y = 1.0f;
  size_t outbase;
  int    outstride;
  if (PASS == 1) {
    int c = line >> 10, y = line & (NPIX - 1);
    const float* src = cube + (size_t)(c * NPIX + y) * NPIX;
    tdm_load_line(bufIn, src, NPIX, 2u);            // 1024 x f32
    invs1y   = invs1[y];
    outbase  = (size_t)c * GSZ * NPIX + (size_t)y;  // + ku*NPIX
    outstride = NPIX;
  } else {
    int c = line >> 11, ku = line & (GSZ - 1);
    const float2* src = cin + (size_t)(c * GSZ + ku) * NPIX;
    tdm_load_line(bufIn, src, NPIX, 3u);            // 1024 x float2
    outbase  = ((size_t)(c * GSZ + ku)) * GSZ;      // + kv
    outstride = 1;
  }
  __builtin_amdgcn_s_wait_tensorcnt(0);             // DMA complete
  asm volatile("" ::: "memory");
  __syncthreads();

  const float* inR = (const float*)bufIn;           // pass-1 view

  // ---- Stage 1: Y[k2][n1] = sum_{n2<16} x[n2*64+n1] * W32^(n2*k2)
  //      (32x16 DFT matrix) x (16x64 data), tiles: M(k2)=2, N(n1)=4, Ksteps=4
  for (int nt = 0; nt < 4; ++nt) {
    for (int mt = 0; mt < 2; ++mt) {
      v8f accr = {0.f,0.f,0.f,0.f,0.f,0.f,0.f,0.f};
      v8f acci = {0.f,0.f,0.f,0.f,0.f,0.f,0.f,0.f};
#pragma unroll
      for (int ks = 0; ks < 4; ++ks) {
        const int m  = mt * 16 + lo;       // k2 row
        const int k0 = ks * 4 + 2 * lh;    // n2 pair (k0, k0+1)
        const float2 a0 = tD32[m * 16 + k0];
        const float2 a1 = tD32[m * 16 + k0 + 1];
        const v2f Ar = { a0.x,  a1.x };
        const v2f Ai = { a0.y,  a1.y };
        const int n1 = nt * 16 + lo;
        if (PASS == 1) {
          // real input with apodization; x index = n2*64 + n1 < 1024 always
          const int x0 = k0 * 64 + n1, x1 = (k0 + 1) * 64 + n1;
          const float b0 = inR[x0] * invs1y * invs1[x0];
          const float b1 = inR[x1] * invs1y * invs1[x1];
          const v2f Br = { b0, b1 };
          accr = WMMA_F32(Ar, Br, accr);
          acci = WMMA_F32(Ai, Br, acci);
        } else {
          const float2 b0 = bufIn[k0 * 64 + n1];
          const float2 b1 = bufIn[(k0 + 1) * 64 + n1];
          const v2f Br  = { b0.x, b1.x };
          const v2f Bi  = { b0.y, b1.y };
          const v2f AiN = { -a0.y, -a1.y };
          accr = WMMA_F32(Ar,  Br, accr);
          accr = WMMA_F32(AiN, Bi, accr);   // Re -= Ai*Bi
          acci = WMMA_F32(Ar,  Bi, acci);
          acci = WMMA_F32(Ai,  Br, acci);
        }
      }
      // twiddle by W2048^(n1*k2), store transposed Z[n1][k2] into LDS
#pragma unroll
      for (int v = 0; v < 8; ++v) {
        const int k2 = mt * 16 + v + 8 * lh;
        const int n1 = nt * 16 + lo;
        const float2 tw = tTW[n1 * 32 + k2];
        const float yr = accr[v], yi = acci[v];
        bufB[n1 * 32 + k2] =
            make_float2(yr * tw.x - yi * tw.y, yr * tw.y + yi * tw.x);
      }
    }
  }
  __syncthreads();

  // ---- Stage 2: X[k2 + 32*k1] = sum_{n1<64} Z[n1][k2] * W64^(n1*k1)
  //      (64x64 DFT matrix) x (64x32 data), tiles: M(k1)=4, N(k2)=2, Ksteps=16
  for (int nt = 0; nt < 2; ++nt) {
    for (int mt = 0; mt < 4; ++mt) {
      v8f accr = {0.f,0.f,0.f,0.f,0.f,0.f,0.f,0.f};
      v8f acci = {0.f,0.f,0.f,0.f,0.f,0.f,0.f,0.f};
#pragma unroll 4
      for (int ks = 0; ks < 16; ++ks) {
        const int m  = mt * 16 + lo;       // k1 row
        const int k0 = ks * 4 + 2 * lh;    // n1 pair
        const float2 a0 = tD64[m * 64 + k0];
        const float2 a1 = tD64[m * 64 + k0 + 1];
        const v2f Ar  = {  a0.x,  a1.x };
        const v2f Ai  = {  a0.y,  a1.y };
        const v2f AiN = { -a0.y, -a1.y };
        const int ncol = nt * 16 + lo;     // k2
        const float2 b0 = bufB[k0 * 32 + ncol];
        const float2 b1 = bufB[(k0 + 1) * 32 + ncol];
        const v2f Br = { b0.x, b1.x };
        const v2f Bi = { b0.y, b1.y };
        accr = WMMA_F32(Ar,  Br, accr);
        accr = WMMA_F32(AiN, Bi, accr);
        acci = WMMA_F32(Ar,  Bi, acci);
        acci = WMMA_F32(Ai,  Br, acci);
      }
#pragma unroll
      for (int v = 0; v < 8; ++v) {
        const int k1 = mt * 16 + v + 8 * lh;
        const int k2 = nt * 16 + lo;
        const int k  = k2 + 32 * k1;       // final frequency index
        cout[outbase + (size_t)k * outstride] =
            make_float2(accr[v], acci[v]);
      }
    }
  }
}

// ---------------- degridding: KB interpolation + phase ----------------
__global__ void nufft_degrid(const float* __restrict__ uu,
                             const float* __restrict__ vv,
                             const float2* __restrict__ ghat_t,  // [c][ku][kv]
                             float* __restrict__ out) {
  const int n = blockIdx.x * blockDim.x + threadIdx.x;
  if (n >= NVIS) return;
  const float DLc = 0.005f * PI_F / 648000.0f;      // cell in radians
  const float KS  = 1000.0f * 2.0f * PI_F * DLc;    // u -> rad/pixel
  const float ku = uu[n] * KS;
  const float kv = vv[n] * KS;
  const float T  = (float)GSZ / (2.0f * PI_F);
  const float tu = ku * T, tv = kv * T;
  const float beta = betaf();

  int   mui[JW], mvi[JW];
  float wu[JW], wv[JW];
  const int fu = (int)floorf(tu), fv = (int)floorf(tv);
#pragma unroll
  for (int j = 0; j < JW; ++j) {
    int mm = fu + (j - (JW / 2 - 1));
    float d = tu - (float)mm;
    float q = 1.0f - (2.0f * d / (float)JW) * (2.0f * d / (float)JW);
    wu[j]  = i0f(beta * sqrtf(fmaxf(q, 0.0f)));
    mui[j] = mm & (GSZ - 1);

    mm = fv + (j - (JW / 2 - 1));
    d = tv - (float)mm;
    q = 1.0f - (2.0f * d / (float)JW) * (2.0f * d / (float)JW);
    wv[j]  = i0f(beta * sqrtf(fmaxf(q, 0.0f)));
    mvi[j] = mm & (GSZ - 1);
  }

  float ps, pc;
  __sincosf((ku + kv) * ((float)NPIX * 0.5f), &ps, &pc);

#pragma unroll
  for (int c = 0; c < NCHAN; ++c) {
    const size_t base = (size_t)c * GSZ * GSZ;
    float ar = 0.f, ai = 0.f;
    for (int j = 0; j < JW; ++j) {
      const size_t cb = base + (size_t)mui[j] * GSZ;
      const float wuj = wu[j];
#pragma unroll
      for (int i = 0; i < JW; ++i) {
        const float2 g = ghat_t[cb + mvi[i]];   // contiguous in kv
        const float w = wuj * wv[i];
        ar += g.x * w;
        ai += g.y * w;
      }
    }
    out[((size_t)c * NVIS + n) * 2]     = ar * pc - ai * ps;
    out[((size_t)c * NVIS + n) * 2 + 1] = ar * ps + ai * pc;
  }
}

extern "C" void kernel_launch(void* const* d_in, const int* in_sizes, int n_in,
                              void* d_out, int out_size, void* d_ws, size_t ws_size,
                              hipStream_t stream) {
  const float* cube = (const float*)d_in[0];
  const float* uu   = (const float*)d_in[1];
  const float* vv   = (const float*)d_in[2];
  float* out = (float*)d_out;

  char* ws = (char*)d_ws;
  float*  invs1 = (float*)(ws + WSO_INVS1);
  float2* d32   = (float2*)(ws + WSO_D32);
  float2* d64   = (float2*)(ws + WSO_D64);
  float2* twd   = (float2*)(ws + WSO_TW);
  float2* r1t   = (float2*)(ws + WSO_R1T);
  float2* ghat  = (float2*)(ws + WSO_GHAT);

  nufft_init_tables<<<8, 256, 0, stream>>>(invs1, d32, d64, twd);
  // Row FFTs: 4 channels x 1024 image rows -> R1t[c][ku][y]
  nufft_fft2048<1><<<NCHAN * NPIX, 32, 0, stream>>>(cube, invs1, d32, d64, twd,
                                                    nullptr, r1t);
  // Column FFTs: 4 channels x 2048 ku columns -> GHAT_T[c][ku][kv]
  nufft_fft2048<2><<<NCHAN * GSZ, 32, 0, stream>>>(cube, invs1, d32, d64, twd,
                                                   r1t, ghat);
  // Degrid 200k visibilities
  nufft_degrid<<<(NVIS + 255) / 256, 256, 0, stream>>>(uu, vv, ghat, out);
}